// Predictor_15066745274464
// MI455X (gfx1250) — compile-verified
//
#include <hip/hip_runtime.h>
#include <hip/hip_bf16.h>
#include <math.h>

typedef __attribute__((ext_vector_type(2))) float v2f;
typedef __attribute__((ext_vector_type(8))) float v8f;

// ---------------------------------------------------------------------------
// V_WMMA_F32_16X16X4_F32 fragment layouts (ISA 7.12.2):
//  A (16x4): lanes 0-15 -> M=lane, VGPR{0,1}=K{0,1}; lanes 16-31 -> K{2,3}.
//  B (4x16): lanes 0-15 -> N=lane, K{0,1}; lanes 16-31 -> N, K{2,3}.
//  C/D (16x16): VGPR i holds row M = i + 8*(lane>>4), col N = lane&15.
// ---------------------------------------------------------------------------

// 16 rows x 64 cols accumulate (max wave-parallelism; used by GRU steps)
__device__ __forceinline__ void wmma_acc_16x64(
    const float* __restrict__ A, int lda,
    const float* __restrict__ W, int ldw, int wcol,
    int K, int arow0, v8f c[4])
{
  const int lane = threadIdx.x & 31;
  const int m    = lane & 15;
  const int kh   = (lane >> 4) << 1;      // 0 or 2
  const float* arow = A + (size_t)(arow0 + m) * lda;
  for (int k = 0; k < K; k += 4) {
    const int kk = k + kh;
    v2f a;
    a.x = arow[kk];
    a.y = arow[kk + 1];
#pragma unroll
    for (int j = 0; j < 4; ++j) {
      const int n = wcol + 16 * j + m;
      v2f b;
      b.x = W[(size_t)kk * ldw + n];
      b.y = W[(size_t)(kk + 1) * ldw + n];
      c[j] = __builtin_amdgcn_wmma_f32_16x16x4_f32(
          false, a, false, b, (short)0, c[j], false, false);
    }
  }
}

// 32 rows x 64 cols accumulate: each B fragment feeds 2 WMMAs
// (8 WMMA : 10 VMEM per k-step; used by the big batched GEMMs)
__device__ __forceinline__ void wmma_acc_32x64(
    const float* __restrict__ A, int lda,
    const float* __restrict__ W, int ldw, int wcol,
    int K, int arow0, v8f c0[4], v8f c1[4])
{
  const int lane = threadIdx.x & 31;
  const int m    = lane & 15;
  const int kh   = (lane >> 4) << 1;
  const float* ar0 = A + (size_t)(arow0 + m) * lda;
  const float* ar1 = ar0 + (size_t)16 * lda;
  for (int k = 0; k < K; k += 4) {
    const int kk = k + kh;
    v2f a0, a1;
    a0.x = ar0[kk]; a0.y = ar0[kk + 1];
    a1.x = ar1[kk]; a1.y = ar1[kk + 1];
#pragma unroll
    for (int j = 0; j < 4; ++j) {
      const int n = wcol + 16 * j + m;
      v2f b;
      b.x = W[(size_t)kk * ldw + n];
      b.y = W[(size_t)(kk + 1) * ldw + n];
      c0[j] = __builtin_amdgcn_wmma_f32_16x16x4_f32(
          false, a0, false, b, (short)0, c0[j], false, false);
      c1[j] = __builtin_amdgcn_wmma_f32_16x16x4_f32(
          false, a1, false, b, (short)0, c1[j], false, false);
    }
  }
}

// ---------------------------------------------------------------------------
// Generic GEMM: C[M,N] = A[M,K] @ W[K,N] (+ bias[N]) (+ residual[M,N])
// One wave per 32x64 output tile. M % 32 == 0, N % 64 == 0, K % 4 == 0.
// ---------------------------------------------------------------------------
__global__ void gemm_bias_res_kernel(
    const float* __restrict__ A, const float* __restrict__ W,
    const float* __restrict__ bias, const float* __restrict__ residual,
    float* __restrict__ C, int M, int N, int K)
{
  const int wave    = (blockIdx.x * blockDim.x + threadIdx.x) >> 5;
  const int tiles_n = N >> 6;
  const int tm      = (wave / tiles_n) << 5;   // 32-row tiles
  const int tn      = (wave % tiles_n) << 6;   // 64-col groups
  if (tm >= M) return;                         // wave-uniform guard

  v8f c0[4], c1[4];
#pragma unroll
  for (int j = 0; j < 4; ++j) {
    c0[j] = (v8f){0.f,0.f,0.f,0.f,0.f,0.f,0.f,0.f};
    c1[j] = (v8f){0.f,0.f,0.f,0.f,0.f,0.f,0.f,0.f};
  }

  wmma_acc_32x64(A, K, W, N, tn, K, tm, c0, c1);

  const int lane = threadIdx.x & 31;
  const int n0   = lane & 15;
  const int mh   = (lane >> 4) * 8;
#pragma unroll
  for (int j = 0; j < 4; ++j) {
    const int   nn = tn + 16 * j + n0;
    const float bv = bias ? bias[nn] : 0.0f;
#pragma unroll
    for (int i = 0; i < 8; ++i) {
      size_t idx = (size_t)(tm + mh + i) * N + nn;
      float v = c0[j][i] + bv;
      if (residual) v += residual[idx];
      C[idx] = v;
      idx += (size_t)16 * N;
      v = c1[j][i] + bv;
      if (residual) v += residual[idx];
      C[idx] = v;
    }
  }
}

// ---------------------------------------------------------------------------
// Action encoder + LayerNorm(ln2), fused per token. One 256-thread block/token.
// ---------------------------------------------------------------------------
__global__ void encoder_ln_kernel(
    const float* __restrict__ action,
    const float* __restrict__ w1, const float* __restrict__ b1,
    const float* __restrict__ w2, const float* __restrict__ b2,
    const float* __restrict__ g,  const float* __restrict__ bln,
    float* __restrict__ kv)
{
  const int tok = blockIdx.x;
  const int t   = threadIdx.x;                 // 0..255
  __shared__ float hid[128];
  __shared__ float red[256];
  const float a0 = action[tok * 2 + 0];
  const float a1 = action[tok * 2 + 1];
  if (t < 128) {
    float h = a0 * w1[t] + a1 * w1[128 + t] + b1[t];
    hid[t] = h > 0.f ? h : 0.f;
  }
  __syncthreads();
  float s = b2[t];
  for (int j = 0; j < 128; ++j) s += hid[j] * w2[j * 256 + t];
  red[t] = s; __syncthreads();
  for (int o = 128; o > 0; o >>= 1) { if (t < o) red[t] += red[t + o]; __syncthreads(); }
  const float mean = red[0] * (1.0f / 256.0f);
  __syncthreads();
  const float d = s - mean;
  red[t] = d * d; __syncthreads();
  for (int o = 128; o > 0; o >>= 1) { if (t < o) red[t] += red[t + o]; __syncthreads(); }
  const float var = red[0] * (1.0f / 256.0f);
  kv[(size_t)tok * 256 + t] = d * rsqrtf(var + 1e-5f) * g[t] + bln[t];
}

// bvo = bv @ wo + bo
__global__ void bvo_kernel(const float* __restrict__ bv, const float* __restrict__ wo,
                           const float* __restrict__ bo, float* __restrict__ bvo)
{
  const int n = threadIdx.x;
  float s = bo[n];
  for (int k = 0; k < 256; ++k) s += bv[k] * wo[k * 256 + n];
  bvo[n] = s;
}

// ---------------------------------------------------------------------------
// GRU per-step GEMM into g_all[512,2048] (16x64 tiles for max parallelism):
//   cols [0,1024):    x_t@wih[:,n] + h@whh[:,n] + bih[n]+bhh[n]   (r,z combined)
//   cols [1024,1536): x_t@wih[:,n] + bih[n]                       (gin)
//   cols [1536,2048): h@whh[:,n-512] + bhh[n-512]                 (ghn)
// ---------------------------------------------------------------------------
__global__ void gru_step_gemm_kernel(
    const float* __restrict__ x_b,    // [512,256]
    const float* __restrict__ h_prev, // [512,512]
    const float* __restrict__ wih,    // [256,1536]
    const float* __restrict__ bih,
    const float* __restrict__ whh,    // [512,1536]
    const float* __restrict__ bhh,
    float* __restrict__ g_all)        // [512,2048]
{
  const int wave = (blockIdx.x * blockDim.x + threadIdx.x) >> 5;
  const int tm   = (wave >> 5) << 4;   // 32 row tiles of 16
  const int tn   = (wave & 31) << 6;   // 32 col groups of 64
  if (tm >= 512) return;

  v8f c[4];
#pragma unroll
  for (int j = 0; j < 4; ++j) c[j] = (v8f){0.f,0.f,0.f,0.f,0.f,0.f,0.f,0.f};

  const int lane = threadIdx.x & 31;
  const int n0   = lane & 15;
  const int mh   = (lane >> 4) * 8;

  if (tn < 1024) {                                   // r,z : K = 256 + 512
    wmma_acc_16x64(x_b,    256, wih, 1536, tn, 256, tm, c);
    wmma_acc_16x64(h_prev, 512, whh, 1536, tn, 512, tm, c);
#pragma unroll
    for (int j = 0; j < 4; ++j) {
      const int nn = tn + 16 * j + n0;
      const float bb = bih[nn] + bhh[nn];
#pragma unroll
      for (int i = 0; i < 8; ++i)
        g_all[(size_t)(tm + mh + i) * 2048 + nn] = c[j][i] + bb;
    }
  } else if (tn < 1536) {                            // gin : from x only
    wmma_acc_16x64(x_b, 256, wih, 1536, tn, 256, tm, c);
#pragma unroll
    for (int j = 0; j < 4; ++j) {
      const int nn = tn + 16 * j + n0;
      const float bb = bih[nn];
#pragma unroll
      for (int i = 0; i < 8; ++i)
        g_all[(size_t)(tm + mh + i) * 2048 + nn] = c[j][i] + bb;
    }
  } else {                                           // ghn : from h only
    wmma_acc_16x64(h_prev, 512, whh, 1536, tn - 512, 512, tm, c);
#pragma unroll
    for (int j = 0; j < 4; ++j) {
      const int nn = tn + 16 * j + n0;
      const float bb = bhh[nn - 512];
#pragma unroll
      for (int i = 0; i < 8; ++i)
        g_all[(size_t)(tm + mh + i) * 2048 + nn] = c[j][i] + bb;
    }
  }
}

// Elementwise GRU gates: h_new = (1-z)*tanh(gin + r*ghn) + z*h_prev
__global__ void gru_gate_kernel(const float* __restrict__ g_all,
                                const float* __restrict__ h_prev,
                                float* __restrict__ h_out)
{
  const int idx = blockIdx.x * blockDim.x + threadIdx.x;   // 512*512
  const int t = idx >> 9, j = idx & 511;
  const float* g = g_all + (size_t)t * 2048;
  const float r = 1.f / (1.f + __expf(-g[j]));
  const float z = 1.f / (1.f + __expf(-g[512 + j]));
  const float n = tanhf(g[1024 + j] + r * g[1536 + j]);
  const float hp = h_prev[idx];
  h_out[idx] = (1.f - z) * n + z * hp;
}

__global__ void zero_kernel(float* __restrict__ p, int n)
{
  const int i = blockIdx.x * blockDim.x + threadIdx.x;
  if (i < n) p[i] = 0.f;
}

// states[:, -1] -> tail of d_out
__global__ void tail_kernel(const float* __restrict__ states, float* __restrict__ tail)
{
  const int i = blockIdx.x * blockDim.x + threadIdx.x;     // 64*256
  const int b = i >> 8, h = i & 255;
  tail[i] = states[((size_t)b * 512 + 511) * 256 + h];
}

// ---------------------------------------------------------------------------
extern "C" void kernel_launch(void* const* d_in, const int* in_sizes, int n_in,
                              void* d_out, int out_size, void* d_ws, size_t ws_size,
                              hipStream_t stream)
{
  const float* state  = (const float*)d_in[0];
  const float* action = (const float*)d_in[1];
  const float* ae_w1  = (const float*)d_in[2];
  const float* ae_b1  = (const float*)d_in[3];
  const float* ae_w2  = (const float*)d_in[4];
  const float* ae_b2  = (const float*)d_in[5];
  // d_in[6..7] = ln1 (dead: feeds only q), d_in[10..13] = wq/bq/wk/bk (dead)
  const float* ln2_g  = (const float*)d_in[8];
  const float* ln2_b  = (const float*)d_in[9];
  const float* wv     = (const float*)d_in[14];
  const float* bv     = (const float*)d_in[15];
  const float* wo     = (const float*)d_in[16];
  const float* bo     = (const float*)d_in[17];
  const float* wih    = (const float*)d_in[18];
  const float* bih    = (const float*)d_in[19];
  const float* whh    = (const float*)d_in[20];
  const float* bhh    = (const float*)d_in[21];
  const float* mlp_w  = (const float*)d_in[22];
  const float* mlp_b  = (const float*)d_in[23];

  const int B = 64, T = 512, H = 256, H2 = 512;
  const int M = B * T;                         // 32768 tokens

  // workspace layout (floats): ~140 MB total
  float* ws   = (float*)d_ws;
  float* kv   = ws;                            // M*H
  float* x    = kv  + (size_t)M * H;           // M*H
  float* outw = x   + (size_t)M * H;           // M*H2  (GRU outputs)
  float* gall = outw + (size_t)M * H2;         // T*2048 (per-step gates)
  float* wvo  = gall + (size_t)T * 2048;       // H*H   (wv@wo folded)
  float* bvo  = wvo + (size_t)H * H;           // H
  float* h0   = bvo + H;                       // T*H2  (zero initial hidden)

  float* states = (float*)d_out;               // [B,T,H]
  float* tail   = states + (size_t)M * H;      // [B,H]

  // h0 = 0
  zero_kernel<<<(T * H2 + 255) / 256, 256, 0, stream>>>(h0, T * H2);

  // kv = LN2(relu(action@w1+b1)@w2+b2)
  encoder_ln_kernel<<<M, 256, 0, stream>>>(action, ae_w1, ae_b1, ae_w2, ae_b2,
                                           ln2_g, ln2_b, kv);

  // fold attention: wvo = wv@wo, bvo = bv@wo + bo
  bvo_kernel<<<1, 256, 0, stream>>>(bv, wo, bo, bvo);
  gemm_bias_res_kernel<<<((H / 32) * (H / 64) * 32) / 256, 256, 0, stream>>>(
      wv, wo, nullptr, nullptr, wvo, H, H, H);

  // x = state + kv@wvo + bvo   (waves = (M/32)*(H/64) = 4096 -> 512 blocks)
  gemm_bias_res_kernel<<<((M / 32) * (H / 64)) / 8, 256, 0, stream>>>(
      kv, wvo, bvo, state, x, M, H, H);

  // GRU recurrence over B (64 sequential steps, T=512 is the batch)
  for (int b = 0; b < B; ++b) {
    const float* hp = (b == 0) ? h0 : (outw + (size_t)(b - 1) * T * H2);
    const float* xb = x + (size_t)b * T * H;
    float*       ho = outw + (size_t)b * T * H2;
    gru_step_gemm_kernel<<<(32 * 32) / 8, 256, 0, stream>>>(
        xb, hp, wih, bih, whh, bhh, gall);
    gru_gate_kernel<<<(T * H2) / 256, 256, 0, stream>>>(gall, hp, ho);
  }

  // states = outw @ mlp_w + mlp_b
  gemm_bias_res_kernel<<<((M / 32) * (H / 64)) / 8, 256, 0, stream>>>(
      outw, mlp_w, mlp_b, nullptr, states, M, H, H2);

  // states[:, -1]
  tail_kernel<<<(B * H) / 256, 256, 0, stream>>>(states, tail);
}